// PyTorchReference_69054484185764
// MI455X (gfx1250) — compile-verified
//
#include <hip/hip_runtime.h>
#include <hip/hip_bf16.h>
#include <stdint.h>

// Bilinear pos-embed gather/interp for MI455X (gfx1250, wave32).
// Memory-bound: ~347MB HBM traffic (weights L2-resident, output streamed NT).
// CDNA5-specific path: async global->LDS staging (ASYNCcnt) double-buffered
// against the blend + non-temporal b128 stores.

typedef __attribute__((ext_vector_type(4))) float f32x4;

#define D_DIM 1280          // embedding dim (static per reference)
#define G_DIM 48            // sqrt(2304)
#define LANES 32            // wave32
#define WAVES_PER_BLOCK 8   // 256 threads
#define NITER (D_DIM / (LANES * 4))   // 10 chunks of 128 floats

__device__ __forceinline__ unsigned lds_off(const void* p) {
    // low 32 bits of a generic pointer to LDS == wave-relative LDS byte offset
    return (unsigned)(uintptr_t)p;
}

__device__ __forceinline__ void async_b128(unsigned lds, uint64_t gaddr) {
    // per-lane 16B  global -> LDS, tracked with ASYNCcnt
    asm volatile("global_load_async_to_lds_b128 %0, %1, off"
                 :: "v"(lds), "v"(gaddr) : "memory");
}

__global__ __launch_bounds__(256)
void pos_embed_kernel(const float* __restrict__ W,
                      const int*   __restrict__ grid,
                      float*       __restrict__ out,
                      int n_entries)
{
    // per-wave: 2 buffers x 4 corners x 32 lanes x float4 = 4KB  (32KB/block)
    __shared__ f32x4 sh[WAVES_PER_BLOCK][2][4][LANES];

    const int wv   = threadIdx.x / LANES;
    const int lane = threadIdx.x % LANES;
    const long long pidx = (long long)blockIdx.x * WAVES_PER_BLOCK + wv;

    // ---- locate (entry, position) by scanning the tiny grid array ----
    long long base_pos = 0, base_row = 0;
    int t = 0, h = 0, w = 0;
    long long pin = -1;
    for (int i = 0; i < n_entries; ++i) {
        const int ti = grid[3*i + 0];
        const int hi = grid[3*i + 1];
        const int wi = grid[3*i + 2];
        const long long np = (long long)hi * wi;
        if (pidx < base_pos + np) { t = ti; h = hi; w = wi; pin = pidx - base_pos; break; }
        base_pos += np;
        base_row += (long long)ti * hi * wi;
    }
    if (pin < 0) return;            // wave beyond total positions (wave-uniform)

    const int y = (int)(pin / w);
    const int x = (int)(pin % w);

    // ---- bilinear corners + weights (matches linspace(0, G-1, n) + trunc) ----
    const float fy = (h > 1) ? (float)y * ((float)(G_DIM - 1) / (float)(h - 1)) : 0.0f;
    const float fx = (w > 1) ? (float)x * ((float)(G_DIM - 1) / (float)(w - 1)) : 0.0f;
    const int hf = (int)fy, wf = (int)fx;
    const int hc = min(hf + 1, G_DIM - 1);
    const int wc = min(wf + 1, G_DIM - 1);
    const float dh = fy - (float)hf, dw = fx - (float)wf;

    const f32x4 w00 = (f32x4)((1.0f - dh) * (1.0f - dw));
    const f32x4 w01 = (f32x4)((1.0f - dh) * dw);
    const f32x4 w10 = (f32x4)(dh * (1.0f - dw));
    const f32x4 w11 = (f32x4)(dh * dw);

    // ---- global byte addresses of the 4 corner rows (lane-offset folded in) ----
    const uint64_t lb = (uint64_t)lane * 16u;
    const uint64_t g0 = (uint64_t)(uintptr_t)(W + (size_t)(hf * G_DIM + wf) * D_DIM) + lb;
    const uint64_t g1 = (uint64_t)(uintptr_t)(W + (size_t)(hf * G_DIM + wc) * D_DIM) + lb;
    const uint64_t g2 = (uint64_t)(uintptr_t)(W + (size_t)(hc * G_DIM + wf) * D_DIM) + lb;
    const uint64_t g3 = (uint64_t)(uintptr_t)(W + (size_t)(hc * G_DIM + wc) * D_DIM) + lb;

    // ---- output row in the spatial-merge-permuted layout ----
    const int hb = y >> 1, mh = y & 1, wb = x >> 1, mw = x & 1;
    const long long crow = ((long long)hb * (w >> 1) + wb) * 4 + mh * 2 + mw;
    float* const outp = out + (base_row + crow) * D_DIM + lane * 4;
    const long long tstride = (long long)h * w * D_DIM;   // rows repeat over t

    const unsigned l0 = lds_off(&sh[wv][0][0][lane]);     // +c*512, +buf*2048

    // ---- prologue prefetch: chunk 0 into buffer 0 ----
    {
        async_b128(l0 +    0, g0);
        async_b128(l0 +  512, g1);
        async_b128(l0 + 1024, g2);
        async_b128(l0 + 1536, g3);
    }

    for (int it = 0; it < NITER; ++it) {
        const int b = it & 1;
        // wait for this buffer's 4 async transfers (only ones outstanding)
        asm volatile("s_wait_asynccnt 0x0" ::: "memory");

        const f32x4 v0 = sh[wv][b][0][lane];
        const f32x4 v1 = sh[wv][b][1][lane];
        const f32x4 v2 = sh[wv][b][2][lane];
        const f32x4 v3 = sh[wv][b][3][lane];
        // ensure LDS reads retired before the buffer can be re-targeted
        asm volatile("s_wait_dscnt 0x0" ::: "memory");

        if (it + 1 < NITER) {       // prefetch next chunk into other buffer
            const unsigned lo = l0 + (unsigned)((b ^ 1) * 2048);
            const uint64_t go = (uint64_t)(it + 1) * 512u;
            async_b128(lo +    0, g0 + go);
            async_b128(lo +  512, g1 + go);
            async_b128(lo + 1024, g2 + go);
            async_b128(lo + 1536, g3 + go);
        }

        // blend (overlaps with the in-flight async loads)
        f32x4 acc = v0 * w00;
        acc += v1 * w01;
        acc += v2 * w10;
        acc += v3 * w11;

        // write once per t-replica, streaming (NT) b128 stores
        float* p = outp + it * (LANES * 4);
        for (int tt = 0; tt < t; ++tt) {
            __builtin_nontemporal_store(acc, (f32x4*)p);
            p += tstride;
        }
    }
}

extern "C" void kernel_launch(void* const* d_in, const int* in_sizes, int n_in,
                              void* d_out, int out_size, void* d_ws, size_t ws_size,
                              hipStream_t stream) {
    const float* W    = (const float*)d_in[0];
    const int*   grid = (const int*)d_in[1];
    float*       out  = (float*)d_out;

    const int n_entries = in_sizes[1] / 3;
    const long long total_rows = (long long)out_size / D_DIM;   // >= total positions (t >= 1)
    const int blocks = (int)((total_rows + WAVES_PER_BLOCK - 1) / WAVES_PER_BLOCK);
    if (blocks > 0) {
        pos_embed_kernel<<<blocks, WAVES_PER_BLOCK * LANES, 0, stream>>>(W, grid, out, n_entries);
    }
}